// Paste2dMulti_26628797236057
// MI455X (gfx1250) — compile-verified
//
#include <hip/hip_runtime.h>

// Paste2dMulti for MI455X (gfx1250, wave32).
//
// block = (canvas strip, image n). 64 KB LDS f32 strip accumulator.
// Per box m:
//   1) async-stage the intersecting image rows (3 channels) into LDS with
//      global_load_async_to_lds_b128 (ASYNCcnt path, coalesced 16B/lane),
//   2) compute each 16x16 mask tile as a rank-1 outer product with
//      V_WMMA_F32_16X16X4_F32 (mask is separable: ymask ⊗ xmask),
//   3) each lane owns 8 pixels of the tile (C/D layout: lane = column,
//      VGPR idx = row), multiplies mask by the channel-summed staged image
//      value and ds_add_f32's into the strip.
// Finally clip to 1.0 and stream out. Strips partition the canvas, so each
// image row is staged exactly once device-wide: ~24 MB read + 16 MB write,
// bandwidth-bound at ~1.7 us on 23.3 TB/s HBM.

#define LCANVAS 256
#define IMH 64
#define IMW 64
#define NCH 3
#define NBOX 8
#define STRIP_H 64
#define NSTRIPS (LCANVAS / STRIP_H)

typedef float v2f __attribute__((ext_vector_type(2)));
typedef float v8f __attribute__((ext_vector_type(8)));

__device__ __forceinline__ float relu6f(float v) {
    return fminf(fmaxf(v, 0.0f), 6.0f);
}

__global__ __launch_bounds__(256) void
paste2d_multi_kernel(const float* __restrict__ coords,
                     const float* __restrict__ images,
                     float* __restrict__ out) {
    __shared__ __align__(16) float strip[STRIP_H * LCANVAS];   // 64 KB
    __shared__ __align__(16) float stage[NCH * IMH * IMW];     // 48 KB

    const int tid      = threadIdx.x;
    const int lane     = tid & 31;
    const int wave     = tid >> 5;        // 0..7
    const int strip_id = blockIdx.x;      // 0..3
    const int n        = blockIdx.y;      // image index
    const int strip_r0 = strip_id * STRIP_H;

    // Low 32 bits of a flat LDS pointer are the LDS byte offset.
    const unsigned stage_lds = (unsigned)(unsigned long long)(&stage[0]);

    // Zero the LDS strip.
    for (int i = tid; i < STRIP_H * LCANVAS; i += 256) strip[i] = 0.0f;
    __syncthreads();

    for (int m = 0; m < NBOX; ++m) {
        const float* c4 = coords + (n * NBOX + m) * 4;
        const float c0 = c4[0], c1 = c4[1], c2 = c4[2], c3 = c4[3];
        // mask ramps: x0 = cx0-1, y0 = cy0-1, x1 = cx1+1, y1 = cy1+1
        const float x0f = c0 - 1.0f, y0f = c1 - 1.0f;
        const float x1f = c2 + 1.0f, y1f = c3 + 1.0f;
        // integer paste offset (coords in [0,256), trunc == floor)
        const int x0i = (int)truncf(c0);
        const int y0i = (int)truncf(c1);

        // Image rows intersecting this strip (block-uniform).
        const int ir_lo = max(0, strip_r0 - y0i);
        const int ir_hi = min(IMH, strip_r0 + STRIP_H - y0i);
        const int cnt   = ir_hi - ir_lo;           // staged row count
        if (cnt <= 0) continue;                    // uniform per block

        const float* imbase =
            images + (size_t)(n * NBOX + m) * NCH * (IMH * IMW);

        // ---- Stage rows [ir_lo, ir_hi) of all 3 channels into LDS ----
        // layout: stage[(ch*cnt + local_r)*64 + col]; one 16B seg per issue.
        {
            const int segs_per_ch = cnt * 16;      // 16B segments per channel
            #pragma unroll
            for (int ch = 0; ch < NCH; ++ch) {
                for (int s = tid; s < segs_per_ch; s += 256) {
                    const int local_r = s >> 4;
                    const int seg     = s & 15;
                    const unsigned voff =
                        (unsigned)(ch * (IMH * IMW * 4) +
                                   (ir_lo + local_r) * (IMW * 4) + seg * 16);
                    const unsigned laddr =
                        stage_lds + (unsigned)((ch * segs_per_ch + s) * 16);
                    asm volatile(
                        "global_load_async_to_lds_b128 %0, %1, %2"
                        :
                        : "v"(laddr), "v"(voff), "s"(imbase)
                        : "memory");
                }
            }
            asm volatile("s_wait_asynccnt 0x0" ::: "memory");
        }
        __syncthreads();

        // ---- 16 tiles of 16x16 cover the 64x64 footprint; 2 per wave ----
        for (int t = wave; t < 16; t += 8) {
            const int ti  = t >> 2;
            const int tj  = t & 3;
            const int tr0 = y0i + ti * 16;   // canvas row of tile origin
            const int tc0 = x0i + tj * 16;   // canvas col of tile origin
            // Wave-uniform skip: tile outside this strip or off-canvas.
            if (tr0 >= strip_r0 + STRIP_H || tr0 + 16 <= strip_r0) continue;
            if (tc0 >= LCANVAS) continue;

            // Rank-1 operands: A[m,0]=ymask(row), B[0,n]=xmask(col), rest 0.
            v2f a; a.x = 0.0f; a.y = 0.0f;
            v2f b; b.x = 0.0f; b.y = 0.0f;
            if (lane < 16) {
                const float gr = (float)(tr0 + lane);
                a.x = relu6f((gr - y0f) * 6.0f) * relu6f((y1f - gr) * 6.0f)
                      * (1.0f / 36.0f);
                const float gc = (float)(tc0 + lane);
                b.x = relu6f((gc - x0f) * 6.0f) * relu6f((x1f - gc) * 6.0f)
                      * (1.0f / 36.0f);
            }
            v8f acc = {};
            acc = __builtin_amdgcn_wmma_f32_16x16x4_f32(
                /*neg_a=*/false, a, /*neg_b=*/false, b,
                /*c_mod=*/(short)0, acc, /*reuse_a=*/false, /*reuse_b=*/false);

            // C/D layout: lane L -> column (L&15); rows = (L>=16)*8 + k.
            const int colc   = tc0 + (lane & 15);
            const bool col_ok = (colc < LCANVAS);
            const int img_c  = col_ok ? (colc - x0i) : 0;     // 0..63
            const int rbase  = tr0 + ((lane >> 4) << 3);
            #pragma unroll
            for (int k = 0; k < 8; ++k) {
                const int rowc = rbase + k;
                const bool ok = col_ok &
                                (rowc >= strip_r0) &
                                (rowc < strip_r0 + STRIP_H) &
                                (rowc < LCANVAS);
                // clamped staged-row index (valid lanes: img_r-ir_lo in [0,cnt))
                int lr = rowc - y0i - ir_lo;
                lr = ok ? lr : 0;
                const int o0 = lr * IMW + img_c;
                const float s = stage[o0] +
                                stage[(cnt * IMW) + o0] +
                                stage[(2 * cnt * IMW) + o0];
                const float val = ok ? (acc[k] * s) : 0.0f;
                // clamped in-strip LDS address; adding 0.0 is harmless
                const int srow = min(max(rowc - strip_r0, 0), STRIP_H - 1);
                const int scol = min(colc, LCANVAS - 1);
                atomicAdd(&strip[srow * LCANVAS + scol], val); // ds_add_f32
            }
        }
        __syncthreads();   // stage reads done before next m overwrites it
    }
    __syncthreads();

    // Clip and stream out (output shape (N,1,256,256)).
    float* o = out + (size_t)n * (LCANVAS * LCANVAS) + (size_t)strip_r0 * LCANVAS;
    for (int i = tid; i < STRIP_H * LCANVAS; i += 256) {
        o[i] = fminf(1.0f, strip[i]);
    }
}

extern "C" void kernel_launch(void* const* d_in, const int* in_sizes, int n_in,
                              void* d_out, int out_size, void* d_ws, size_t ws_size,
                              hipStream_t stream) {
    const float* coords = (const float*)d_in[0];
    const float* images = (const float*)d_in[1];
    float* out = (float*)d_out;

    const int N = in_sizes[0] / (NBOX * 4);   // 64 for the reference sizes
    dim3 grid(NSTRIPS, N);
    paste2d_multi_kernel<<<grid, 256, 0, stream>>>(coords, images, out);
}